// TransitionUp_3375844295200
// MI455X (gfx1250) — compile-verified
//
#include <hip/hip_runtime.h>
#include <hip/hip_bf16.h>

typedef __attribute__((ext_vector_type(2))) float v2f;
typedef __attribute__((ext_vector_type(8))) float v8f;

// Gate strictly on the device compilation pass.
#if defined(__HIP_DEVICE_COMPILE__) && !__has_builtin(__builtin_amdgcn_wmma_f32_16x16x4_f32)
#error "gfx1250 device pass: missing __builtin_amdgcn_wmma_f32_16x16x4_f32"
#endif

// ---------------------------------------------------------------------------
// Generic GEMM + bias:  C[M,N] = A[M,K] @ B[K,N] + bias[N]   (all fp32)
// Tile: 64(M) x 128(N) x 32(K). 256 threads = 8 waves (wave32).
// Wave grid 4(M) x 2(N); each wave owns 16 rows x 64 cols = 4 WMMA accums.
// LDS: A row-major stride 36 dwords, B stored transposed [n][k] stride 36
// -> conflict-free ds_load_b64 operand fetch, 8B-aligned.
// ---------------------------------------------------------------------------
#define TILE_M 64
#define TILE_N 128
#define TILE_K 32
#define A_STR 36
#define B_STR 36

__global__ __launch_bounds__(256)
void gemm_bias_kernel(const float* __restrict__ A, const float* __restrict__ B,
                      const float* __restrict__ bias, float* __restrict__ C,
                      int M, int N, int K) {
    __shared__ float sA[TILE_M * A_STR];   //  9216 B
    __shared__ float sB[TILE_N * B_STR];   // 18432 B

    const int tid  = threadIdx.x;
    const int lane = tid & 31;
    const int wave = tid >> 5;
    const int wm   = wave & 3;        // M sub-tile 0..3
    const int wn   = wave >> 2;       // N sub-tile 0..1 (64 cols each)
    const int lrow = lane & 15;       // row (A) / col (B,C/D) within 16
    const int hi   = lane >> 4;       // half-wave select

    const int m0 = blockIdx.x * TILE_M;
    const int n0 = blockIdx.y * TILE_N;

    v8f acc[4];
#pragma unroll
    for (int i = 0; i < 4; ++i)
#pragma unroll
        for (int j = 0; j < 8; ++j) acc[i][j] = 0.0f;

    for (int k0 = 0; k0 < K; k0 += TILE_K) {
        // ---- stage A tile 64x32 (2 float4 / thread) ----
#pragma unroll
        for (int q = 0; q < 2; ++q) {
            int lin = tid + q * 256;          // float4 id, 512 total
            int r   = lin >> 3;               // 8 float4 per 32-col row
            int c4  = lin & 7;
            float4 v = *(const float4*)(A + (size_t)(m0 + r) * K + k0 + c4 * 4);
            *(float4*)&sA[r * A_STR + c4 * 4] = v;
        }
        // ---- stage B tile 32x128 transposed (4 float4 / thread) ----
#pragma unroll
        for (int q = 0; q < 4; ++q) {
            int lin = tid + q * 256;          // float4 id, 1024 total
            int r   = lin >> 5;               // k row 0..31
            int c4  = lin & 31;
            float4 v = *(const float4*)(B + (size_t)(k0 + r) * N + n0 + c4 * 4);
            sB[(c4 * 4 + 0) * B_STR + r] = v.x;
            sB[(c4 * 4 + 1) * B_STR + r] = v.y;
            sB[(c4 * 4 + 2) * B_STR + r] = v.z;
            sB[(c4 * 4 + 3) * B_STR + r] = v.w;
        }
        // prefetch next K-tile while this one is consumed
        if (k0 + TILE_K < K) {
            __builtin_prefetch(A + (size_t)(m0 + (tid >> 3)) * K + k0 + TILE_K, 0, 1);
            __builtin_prefetch(B + (size_t)(k0 + TILE_K + (tid >> 5)) * N + n0 + (tid & 31) * 4, 0, 1);
        }
        __syncthreads();

        // ---- 8 k-steps of v_wmma_f32_16x16x4_f32 per accumulator ----
#pragma unroll
        for (int kk = 0; kk < 8; ++kk) {
            // A 16x4: lanes0-15 hold K={0,1}, lanes16-31 hold K={2,3}
            v2f a = *(const v2f*)&sA[(wm * 16 + lrow) * A_STR + kk * 4 + hi * 2];
#pragma unroll
            for (int nt = 0; nt < 4; ++nt) {
                v2f b = *(const v2f*)&sB[(wn * 64 + nt * 16 + lrow) * B_STR + kk * 4 + hi * 2];
                acc[nt] = __builtin_amdgcn_wmma_f32_16x16x4_f32(
                    false, a, false, b, (short)0, acc[nt], false, false);
            }
        }
        __syncthreads();
    }

    // ---- writeback: VGPR i -> row i (lanes 0-15) / row i+8 (lanes 16-31) ----
#pragma unroll
    for (int nt = 0; nt < 4; ++nt) {
        int col = n0 + wn * 64 + nt * 16 + lrow;
        float bv = bias[col];
#pragma unroll
        for (int i = 0; i < 8; ++i) {
            int row = m0 + wm * 16 + i + hi * 8;
            C[(size_t)row * N + col] = acc[nt][i] + bv;
        }
    }
}

// ---------------------------------------------------------------------------
// Deterministic per-column batch stats: mean + rsqrt(var+eps)
// One block handles 16 columns; 16 row-threads per column, fixed LDS tree.
// ---------------------------------------------------------------------------
__global__ __launch_bounds__(256)
void col_stats_kernel(const float* __restrict__ H, float* __restrict__ mean_out,
                      float* __restrict__ invstd_out, int M, int N) {
    __shared__ float s_sum[256];
    __shared__ float s_sq[256];
    const int c  = threadIdx.x & 15;
    const int r0 = threadIdx.x >> 4;
    const int cg = blockIdx.x * 16;
    float sum = 0.f, sq = 0.f;
    for (int r = r0; r < M; r += 16) {
        float v = H[(size_t)r * N + cg + c];
        sum += v;
        sq  += v * v;
    }
    s_sum[threadIdx.x] = sum;
    s_sq[threadIdx.x]  = sq;
    __syncthreads();
    for (int s = 8; s > 0; s >>= 1) {
        if (r0 < s) {
            s_sum[r0 * 16 + c] += s_sum[(r0 + s) * 16 + c];
            s_sq[r0 * 16 + c]  += s_sq[(r0 + s) * 16 + c];
        }
        __syncthreads();
    }
    if (r0 == 0) {
        float m   = s_sum[c] / (float)M;
        float var = s_sq[c] / (float)M - m * m;
        mean_out[cg + c]   = m;
        invstd_out[cg + c] = rsqrtf(var + 1e-5f);
    }
}

// ---------------------------------------------------------------------------
// Elementwise BN + ReLU
// ---------------------------------------------------------------------------
__global__ __launch_bounds__(256)
void bn_relu_kernel(const float* __restrict__ H, const float* __restrict__ mean,
                    const float* __restrict__ invstd, const float* __restrict__ gamma,
                    const float* __restrict__ beta, float* __restrict__ O,
                    int total, int N) {
    int i = blockIdx.x * 256 + threadIdx.x;
    if (i < total) {
        int c = i % N;
        float v = (H[i] - mean[c]) * invstd[c] * gamma[c] + beta[c];
        O[i] = v > 0.f ? v : 0.f;
    }
}

// ---------------------------------------------------------------------------
// kNN (k=3): coarse positions staged in LDS (48 KB @ 4096 pts), each thread
// scans all coarse points for one fine point. Strict < insertion preserves
// lowest-index-first tie ordering, matching lax.top_k stability.
// ---------------------------------------------------------------------------
__global__ __launch_bounds__(256)
void knn_kernel(const float* __restrict__ pos, const float* __restrict__ pos_sub,
                int* __restrict__ idx_out, float* __restrict__ w_out,
                int Npts, int Nsub) {
    extern __shared__ float spos[];
    for (int i = threadIdx.x; i < Nsub * 3; i += 256) spos[i] = pos_sub[i];
    __syncthreads();
    int n = blockIdx.x * 256 + threadIdx.x;
    if (n >= Npts) return;
    const float px = pos[n * 3 + 0], py = pos[n * 3 + 1], pz = pos[n * 3 + 2];
    float d0 = 3.4e38f, d1 = 3.4e38f, d2 = 3.4e38f;
    int   i0 = 0, i1 = 0, i2 = 0;
    for (int j = 0; j < Nsub; ++j) {
        float dx = px - spos[j * 3 + 0];
        float dy = py - spos[j * 3 + 1];
        float dz = pz - spos[j * 3 + 2];
        float d  = dx * dx + dy * dy + dz * dz;
        if (d < d2) {
            if (d < d1) {
                if (d < d0) { d2 = d1; i2 = i1; d1 = d0; i1 = i0; d0 = d; i0 = j; }
                else        { d2 = d1; i2 = i1; d1 = d;  i1 = j; }
            } else          { d2 = d;  i2 = j; }
        }
    }
    idx_out[n * 3 + 0] = i0;
    idx_out[n * 3 + 1] = i1;
    idx_out[n * 3 + 2] = i2;
    w_out[n * 3 + 0] = 1.f / fmaxf(d0, 1e-16f);
    w_out[n * 3 + 1] = 1.f / fmaxf(d1, 1e-16f);
    w_out[n * 3 + 2] = 1.f / fmaxf(d2, 1e-16f);
}

// ---------------------------------------------------------------------------
// Fused epilogue: out = BN_ReLU(h2_pre) + IDW-interp(h_sub). One block/point.
// ---------------------------------------------------------------------------
__global__ __launch_bounds__(128)
void final_kernel(const float* __restrict__ H2, const float* __restrict__ hsub,
                  const float* __restrict__ mean, const float* __restrict__ invstd,
                  const float* __restrict__ gamma, const float* __restrict__ beta,
                  const int* __restrict__ knn_idx, const float* __restrict__ knn_w,
                  float* __restrict__ out, int N) {
    const int n = blockIdx.x;
    const int j0 = knn_idx[n * 3 + 0], j1 = knn_idx[n * 3 + 1], j2 = knn_idx[n * 3 + 2];
    const float w0 = knn_w[n * 3 + 0], w1 = knn_w[n * 3 + 1], w2 = knn_w[n * 3 + 2];
    const float inv_den = 1.f / (w0 + w1 + w2);
    const float* r0 = hsub + (size_t)j0 * N;
    const float* r1 = hsub + (size_t)j1 * N;
    const float* r2 = hsub + (size_t)j2 * N;
    for (int c = threadIdx.x; c < N; c += 128) {
        float h = (H2[(size_t)n * N + c] - mean[c]) * invstd[c] * gamma[c] + beta[c];
        h = fmaxf(h, 0.f);
        float interp = (w0 * r0[c] + w1 * r1[c] + w2 * r2[c]) * inv_den;
        out[(size_t)n * N + c] = h + interp;
    }
}

// ---------------------------------------------------------------------------
extern "C" void kernel_launch(void* const* d_in, const int* in_sizes, int n_in,
                              void* d_out, int out_size, void* d_ws, size_t ws_size,
                              hipStream_t stream) {
    const float* x        = (const float*)d_in[0];
    const float* x_sub    = (const float*)d_in[1];
    const float* pos      = (const float*)d_in[2];
    const float* pos_sub  = (const float*)d_in[3];
    const float* W_sub    = (const float*)d_in[4];
    const float* b_sub    = (const float*)d_in[5];
    const float* gamma_s  = (const float*)d_in[6];
    const float* beta_s   = (const float*)d_in[7];
    const float* W        = (const float*)d_in[8];
    const float* b        = (const float*)d_in[9];
    const float* gamma    = (const float*)d_in[10];
    const float* beta     = (const float*)d_in[11];

    const int C    = in_sizes[5];           // 512  (C_OUT, from b_sub)
    const int Nsub = in_sizes[3] / 3;       // 4096
    const int Npts = in_sizes[2] / 3;       // 16384
    const int Cin  = in_sizes[1] / Nsub;    // 1024

    // workspace layout (floats)
    float* ws      = (float*)d_ws;
    float* h1      = ws;                              // Nsub*C
    float* hsub    = h1 + (size_t)Nsub * C;           // Nsub*C
    float* h2      = hsub + (size_t)Nsub * C;         // Npts*C
    float* mean1   = h2 + (size_t)Npts * C;           // C
    float* invst1  = mean1 + C;                       // C
    float* mean2   = invst1 + C;                      // C
    float* invst2  = mean2 + C;                       // C
    float* knn_w   = invst2 + C;                      // Npts*3
    int*   knn_idx = (int*)(knn_w + (size_t)Npts * 3);// Npts*3

    // 1) coarse GEMM: h1 = x_sub @ W_sub + b_sub   [Nsub, C]
    gemm_bias_kernel<<<dim3(Nsub / TILE_M, C / TILE_N), 256, 0, stream>>>(
        x_sub, W_sub, b_sub, h1, Nsub, C, Cin);
    // 2) BN stats over coarse batch
    col_stats_kernel<<<C / 16, 256, 0, stream>>>(h1, mean1, invst1, Nsub, C);
    // 3) h_sub = BN+ReLU
    bn_relu_kernel<<<((size_t)Nsub * C + 255) / 256, 256, 0, stream>>>(
        h1, mean1, invst1, gamma_s, beta_s, hsub, Nsub * C, C);
    // 4) kNN(3) indices + inverse-square-distance weights
    knn_kernel<<<(Npts + 255) / 256, 256, (size_t)Nsub * 3 * sizeof(float), stream>>>(
        pos, pos_sub, knn_idx, knn_w, Npts, Nsub);
    // 5) fine GEMM: h2 = x @ W + b   [Npts, C]
    gemm_bias_kernel<<<dim3(Npts / TILE_M, C / TILE_N), 256, 0, stream>>>(
        x, W, b, h2, Npts, C, C);
    // 6) BN stats over fine batch
    col_stats_kernel<<<C / 16, 256, 0, stream>>>(h2, mean2, invst2, Npts, C);
    // 7) fused BN+ReLU + interpolation add
    final_kernel<<<Npts, 128, 0, stream>>>(
        h2, hsub, mean2, invst2, gamma, beta, knn_idx, knn_w, (float*)d_out, C);
}